// EnsembleRatioModel_72146860638430
// MI455X (gfx1250) — compile-verified
//
#include <hip/hip_runtime.h>
#include <hip/hip_bf16.h>
#include <math.h>

// fp32 WMMA fragment types (CDNA5 gfx1250, wave32)
typedef __attribute__((ext_vector_type(2))) float v2f;   // A/B frag: 16x4 f32 = 2 VGPRs
typedef __attribute__((ext_vector_type(8))) float v8f;   // C/D frag: 16x16 f32 = 8 VGPRs
typedef __attribute__((ext_vector_type(4))) int   v4i;   // 16B transfer unit

#define TM 128   // block tile M
#define TN 128   // block tile N
#define TK 16    // K chunk staged in LDS
#define LDA 20   // padded LDS stride for A tile (floats); 80B keeps 16B alignment
#define LDB 132  // padded LDS stride for B tile (floats); 528B keeps 16B alignment

// Row chunk so that producer/consumer activations stay resident in the 192MB L2:
// 32768 rows x 512 cols x 4B = 64MB per buffer; peak working set ~160MB.
#define ROWS_PER_CHUNK 32768

// Async global->LDS path (CDNA5).
#if defined(__has_builtin)
#if __has_builtin(__builtin_amdgcn_global_load_async_to_lds_b128) && \
    __has_builtin(__builtin_amdgcn_s_wait_asynccnt)
#define USE_ASYNC_LDS 1
#endif
#endif

#ifdef USE_ASYNC_LDS
typedef __attribute__((address_space(1))) v4i gas_v4i;
typedef __attribute__((address_space(3))) v4i las_v4i;
#define GPTR(p) ((gas_v4i*)(p))
#define LPTR(p) ((las_v4i*)(p))
#endif

// C[M,Nn] = act_out( act_in(A[M,K]) @ B[K,Nn] + bias[Nn] )
//   RELU_IN:  apply relu to A at fragment read (only for external inputs x / repr)
//   RELU_OUT: apply relu at store (intermediates are stored post-relu)
// 256 threads = 8 waves arranged 4(M) x 2(N); each wave computes 32x64 of C
// as 2x4 fp32 WMMA accumulators. Double-buffered LDS, pipelined K loop.
template <bool RELU_IN, bool RELU_OUT>
__global__ __launch_bounds__(256)
void gemm_wmma(const float* __restrict__ A,
               const float* __restrict__ B,
               const float* __restrict__ bias,
               float* __restrict__ C,
               int M, int Nn, int K) {
    __shared__ float As[2][TM * LDA];
    __shared__ float Bs[2][TK * LDB];

    const int tid  = threadIdx.x;
    const int lane = tid & 31;
    const int wave = tid >> 5;     // 0..7
    const int wm   = wave >> 1;    // 0..3  (M direction, 32 rows each)
    const int wn   = wave & 1;     // 0..1  (N direction, 64 cols each)

    const int blockM = blockIdx.y * TM;
    const int blockN = blockIdx.x * TN;

    const v8f zero8 = {0.f, 0.f, 0.f, 0.f, 0.f, 0.f, 0.f, 0.f};
    v8f acc[2][4];
    #pragma unroll
    for (int mt = 0; mt < 2; ++mt)
        #pragma unroll
        for (int nt = 0; nt < 4; ++nt)
            acc[mt][nt] = zero8;

    // Global->LDS staging map: each thread moves 8 floats of A and 8 of B per chunk.
    const int arow  = tid >> 1;            // 0..127
    const int acolb = (tid & 1) << 3;      // 0 or 8
    const int brow  = tid >> 4;            // 0..15
    const int bcolb = (tid & 15) << 3;     // 0..120

    const float* Ag = A + (size_t)(blockM + arow) * K + acolb;
    const float* Bg = B + (size_t)brow * Nn + (blockN + bcolb);
    const int aoff = arow * LDA + acolb;
    const int boff = brow * LDB + bcolb;

    const int nchunks = K / TK;

#ifdef USE_ASYNC_LDS
    {
        __builtin_amdgcn_global_load_async_to_lds_b128(GPTR(Ag),     LPTR(&As[0][aoff]),     0, 0);
        __builtin_amdgcn_global_load_async_to_lds_b128(GPTR(Ag + 4), LPTR(&As[0][aoff + 4]), 0, 0);
        __builtin_amdgcn_global_load_async_to_lds_b128(GPTR(Bg),     LPTR(&Bs[0][boff]),     0, 0);
        __builtin_amdgcn_global_load_async_to_lds_b128(GPTR(Bg + 4), LPTR(&Bs[0][boff + 4]), 0, 0);
        __builtin_amdgcn_s_wait_asynccnt(0);
    }
    __syncthreads();
#else
    {
        const float4 ra0 = *(const float4*)(Ag);
        const float4 ra1 = *(const float4*)(Ag + 4);
        const float4 rb0 = *(const float4*)(Bg);
        const float4 rb1 = *(const float4*)(Bg + 4);
        *(float4*)&As[0][aoff]     = ra0;
        *(float4*)&As[0][aoff + 4] = ra1;
        *(float4*)&Bs[0][boff]     = rb0;
        *(float4*)&Bs[0][boff + 4] = rb1;
    }
    __syncthreads();
#endif

    for (int c = 0; c < nchunks; ++c) {
        const int cur = c & 1;
        const int nxt = cur ^ 1;
        const bool has_next = (c + 1) < nchunks;

#ifdef USE_ASYNC_LDS
        if (has_next) {
            const float* ga = Ag + (c + 1) * TK;
            const float* gb = Bg + (size_t)(c + 1) * TK * Nn;
            __builtin_amdgcn_global_load_async_to_lds_b128(GPTR(ga),     LPTR(&As[nxt][aoff]),     0, 0);
            __builtin_amdgcn_global_load_async_to_lds_b128(GPTR(ga + 4), LPTR(&As[nxt][aoff + 4]), 0, 0);
            __builtin_amdgcn_global_load_async_to_lds_b128(GPTR(gb),     LPTR(&Bs[nxt][boff]),     0, 0);
            __builtin_amdgcn_global_load_async_to_lds_b128(GPTR(gb + 4), LPTR(&Bs[nxt][boff + 4]), 0, 0);
        }
#else
        float4 ra0, ra1, rb0, rb1;
        if (has_next) {
            const float* ga = Ag + (c + 1) * TK;
            const float* gb = Bg + (size_t)(c + 1) * TK * Nn;
            ra0 = *(const float4*)(ga);
            ra1 = *(const float4*)(ga + 4);
            rb0 = *(const float4*)(gb);
            rb1 = *(const float4*)(gb + 4);
        }
#endif

        // ---- compute chunk c from LDS buffer `cur` ----
        #pragma unroll
        for (int kk = 0; kk < TK / 4; ++kk) {
            // ISA 16x4 f32 A layout: M = lane%16; VGPR0/1 hold K = 2*(lane/16), +1
            const int akb = kk * 4 + ((lane >> 4) << 1);

            v2f afr[2];
            #pragma unroll
            for (int mt = 0; mt < 2; ++mt) {
                const int base = (wm * 32 + mt * 16 + (lane & 15)) * LDA + akb;
                float a0 = As[cur][base];
                float a1 = As[cur][base + 1];
                if (RELU_IN) { a0 = fmaxf(a0, 0.f); a1 = fmaxf(a1, 0.f); }
                afr[mt].x = a0;
                afr[mt].y = a1;
            }
            v2f bfr[4];
            #pragma unroll
            for (int nt = 0; nt < 4; ++nt) {
                const int cn = wn * 64 + nt * 16 + (lane & 15);
                bfr[nt].x = Bs[cur][akb * LDB + cn];
                bfr[nt].y = Bs[cur][(akb + 1) * LDB + cn];
            }
            #pragma unroll
            for (int mt = 0; mt < 2; ++mt)
                #pragma unroll
                for (int nt = 0; nt < 4; ++nt)
                    acc[mt][nt] = __builtin_amdgcn_wmma_f32_16x16x4_f32(
                        false, afr[mt], false, bfr[nt], (short)0, acc[mt][nt], false, false);
        }

        if (has_next) {
#ifdef USE_ASYNC_LDS
            __builtin_amdgcn_s_wait_asynccnt(0);
#else
            *(float4*)&As[nxt][aoff]     = ra0;
            *(float4*)&As[nxt][aoff + 4] = ra1;
            *(float4*)&Bs[nxt][boff]     = rb0;
            *(float4*)&Bs[nxt][boff + 4] = rb1;
#endif
            __syncthreads();
        }
    }

    // C/D layout: VGPR r -> M = r (lanes 0-15) or r+8 (lanes 16-31); N = lane%16
    const int rhi  = (lane >> 4) << 3;
    const int ncol = lane & 15;
    #pragma unroll
    for (int mt = 0; mt < 2; ++mt) {
        #pragma unroll
        for (int nt = 0; nt < 4; ++nt) {
            const int col = blockN + wn * 64 + nt * 16 + ncol;
            const float bb = bias[col];
            #pragma unroll
            for (int r = 0; r < 8; ++r) {
                const int row = blockM + wm * 32 + mt * 16 + rhi + r;
                float v = acc[mt][nt][r] + bb;
                if (RELU_OUT) v = fmaxf(v, 0.f);
                C[(size_t)row * Nn + col] = v;
            }
        }
    }
}

// Per-pair head: logit = S1relu[n,:] . w + b (S1 stored post-relu); sigmoid/clip/ratio/mask.
// One wave per event row; 8 rows per 256-thread block.
__global__ __launch_bounds__(256)
void head_kernel(const float* __restrict__ S1, const float* __restrict__ w,
                 const float* __restrict__ b, const int* __restrict__ y,
                 const int* __restrict__ pairs, int p, int Kdim, int N,
                 float* __restrict__ r_out, float* __restrict__ s_out,
                 float* __restrict__ m_out) {
    const int wave = threadIdx.x >> 5;
    const int lane = threadIdx.x & 31;
    const int row  = blockIdx.x * 8 + wave;
    if (row >= N) return;

    const float* s1 = S1 + (size_t)row * Kdim;
    float sum = 0.f;
    for (int k = lane; k < Kdim; k += 32)
        sum += s1[k] * w[k];
    #pragma unroll
    for (int off = 16; off > 0; off >>= 1)
        sum += __shfl_xor(sum, off, 32);

    if (lane == 0) {
        const float logit = sum + b[0];
        float s = 1.f / (1.f + expf(-logit));
        s = fmaxf(s, 1e-9f);
        const float r  = (1.f - s) / s;
        const int   yv = y[row];
        const float m  = (yv == pairs[2 * p] || yv == pairs[2 * p + 1]) ? 1.f : 0.f;
        r_out[row] = r * m;
        s_out[row] = s * m;
        m_out[row] = m;
    }
}

extern "C" void kernel_launch(void* const* d_in, const int* in_sizes, int n_in,
                              void* d_out, int out_size, void* d_ws, size_t ws_size,
                              hipStream_t stream) {
    const float* x     = (const float*)d_in[0];
    const int*   y     = (const int*)d_in[1];
    const int*   pairs = (const int*)d_in[2];
    const float* W0 = (const float*)d_in[3];
    const float* b0 = (const float*)d_in[4];
    const float* W1 = (const float*)d_in[5];
    const float* b1 = (const float*)d_in[6];
    const float* W2 = (const float*)d_in[7];
    const float* b2 = (const float*)d_in[8];
    const float* SW0 = (const float*)d_in[9];
    const float* Sb0 = (const float*)d_in[10];
    const float* SW1 = (const float*)d_in[11];
    const float* Sb1 = (const float*)d_in[12];
    const float* SW2 = (const float*)d_in[13];
    const float* Sb2 = (const float*)d_in[14];

    const int N    = in_sizes[1];             // 131072
    const int NOBS = in_sizes[0] / N;         // 64
    const int H0   = in_sizes[4];             // 512
    const int H1   = in_sizes[6];             // 512
    const int H2   = in_sizes[8];             // 512
    const int P    = in_sizes[2] / 2;         // 3
    const int S0d  = in_sizes[10] / P;        // 512
    const int S1d  = in_sizes[12] / P;        // 256

    float* reprO = (float*)d_out;                         // [N, H2] (pre-relu, as reference)
    float* r_out = reprO + (size_t)N * H2;                // [P, N]
    float* s_out = r_out + (size_t)P * N;                 // [P, N]
    float* m_out = s_out + (size_t)P * N;                 // [P, N]

    float* buf0 = (float*)d_ws;                           // [ROWS_PER_CHUNK, 512]
    float* buf1 = buf0 + (size_t)ROWS_PER_CHUNK * H0;     // [ROWS_PER_CHUNK, 512]

    dim3 blk(256);

    // Process events in L2-resident row chunks: the whole layer stack runs per
    // chunk so every intermediate activation is consumed out of L2, not HBM.
    for (int cs = 0; cs < N; cs += ROWS_PER_CHUNK) {
        const int ch = (N - cs < ROWS_PER_CHUNK) ? (N - cs) : ROWS_PER_CHUNK;
        const float* xc   = x + (size_t)cs * NOBS;
        float*       repc = reprO + (size_t)cs * H2;

        // Trunk: buf0 <- relu(h0), buf1 <- relu(h1), repr <- pre-relu
        gemm_wmma<true,  true ><<<dim3(H0 / TN, ch / TM), blk, 0, stream>>>(xc,   W0, b0, buf0, ch, H0, NOBS);
        gemm_wmma<false, true ><<<dim3(H1 / TN, ch / TM), blk, 0, stream>>>(buf0, W1, b1, buf1, ch, H1, H0);
        gemm_wmma<false, false><<<dim3(H2 / TN, ch / TM), blk, 0, stream>>>(buf1, W2, b2, repc, ch, H2, H1);

        // Per-pair subnetworks: relu(repr) on input; intermediates post-relu.
        for (int p = 0; p < P; ++p) {
            gemm_wmma<true,  true><<<dim3(S0d / TN, ch / TM), blk, 0, stream>>>(
                repc, SW0 + (size_t)p * H2 * S0d, Sb0 + (size_t)p * S0d, buf0, ch, S0d, H2);
            gemm_wmma<false, true><<<dim3(S1d / TN, ch / TM), blk, 0, stream>>>(
                buf0, SW1 + (size_t)p * S0d * S1d, Sb1 + (size_t)p * S1d, buf1, ch, S1d, S0d);
            head_kernel<<<dim3((ch + 7) / 8), blk, 0, stream>>>(
                buf1, SW2 + (size_t)p * S1d, Sb2 + p, y + cs, pairs, p, S1d, ch,
                r_out + (size_t)p * N + cs, s_out + (size_t)p * N + cs, m_out + (size_t)p * N + cs);
        }
    }
}